// LNRNet_3676492006253
// MI455X (gfx1250) — compile-verified
//
#include <hip/hip_runtime.h>

// Problem constants (match reference):  B=8, C=8*16+70=198, H=W=256, J=16
#define NB   8
#define NJ   16
#define NP   65536      // H*W
#define NCH  198
#define CH_MASK 3
#define CH_LOC  4
#define CH_POSE 52
#define CH_SKIN 100
#define CH_CONF 118
#define NCHUNK  64      // K-split chunks per (b,j) in the vote reduction

typedef __attribute__((ext_vector_type(2))) float v2f;
typedef __attribute__((ext_vector_type(8))) float v8f;

__device__ __forceinline__ float sigm(float x) {
  return 1.0f / (1.0f + __expf(-x));
}

// ---------------------------------------------------------------------------
// Kernel 1: voting partial reduction.
// grid = (NB*NJ, NCHUNK), block = 256 (8 waves of 32).
// Each block accumulates 7 quantities over its pixel chunk, then reduces:
//   per-wave shuffle tree -> 8 wave sums in LDS -> two chained
//   v_wmma_f32_16x16x4_f32 with an all-ones B matrix sum the 8 wave partials
//   (K slots = waves, M rows = quantities). Partial written to ws (no atomics
//   => deterministic).
// ---------------------------------------------------------------------------
__global__ void vote_kernel(const float* __restrict__ seg,
                            float* __restrict__ part) {
  const int bj = blockIdx.x;          // 0..127
  const int b  = bj >> 4;
  const int j  = bj & 15;
  const size_t base = (size_t)b * NCH * NP;
  const float* __restrict__ maskp = seg + base + (size_t)CH_MASK * NP;
  const float* __restrict__ confp = seg + base + (size_t)(CH_CONF + j) * NP;
  const float* __restrict__ locp  = seg + base + (size_t)(CH_LOC  + 3 * j) * NP;
  const float* __restrict__ posep = seg + base + (size_t)(CH_POSE + 3 * j) * NP;

  float a0 = 0.f, a1 = 0.f, a2 = 0.f, a3 = 0.f, a4 = 0.f, a5 = 0.f, a6 = 0.f;
  const int tid    = threadIdx.x;
  const int stride = blockDim.x * gridDim.y;          // 256*64 = 16384
  for (int p = blockIdx.y * blockDim.x + tid; p < NP; p += stride) {
    float m = sigm(maskp[p]);          // mask = sigmoid(ch3)
    float s = m * sigm(confp[p]);      // score = sigmoid(conf)*mask
    float w = s * m;                   // jm carries one more mask factor
    a0 += s;
    a1 += locp[p]           * w;
    a2 += locp[p + NP]      * w;
    a3 += locp[p + 2 * NP]  * w;
    a4 += posep[p]          * w;
    a5 += posep[p + NP]     * w;
    a6 += posep[p + 2 * NP] * w;
  }

  // Intra-wave reduction (wave32).
  #pragma unroll
  for (int off = 16; off > 0; off >>= 1) {
    a0 += __shfl_xor(a0, off, 32);
    a1 += __shfl_xor(a1, off, 32);
    a2 += __shfl_xor(a2, off, 32);
    a3 += __shfl_xor(a3, off, 32);
    a4 += __shfl_xor(a4, off, 32);
    a5 += __shfl_xor(a5, off, 32);
    a6 += __shfl_xor(a6, off, 32);
  }

  __shared__ float wsum[8][16];   // [wave][quantity], quantities 7..15 zero
  const int wave = tid >> 5;
  const int lane = tid & 31;
  if (lane == 0) {
    wsum[wave][0] = a0; wsum[wave][1] = a1; wsum[wave][2] = a2;
    wsum[wave][3] = a3; wsum[wave][4] = a4; wsum[wave][5] = a5;
    wsum[wave][6] = a6;
    #pragma unroll
    for (int q = 7; q < 16; ++q) wsum[wave][q] = 0.0f;
  }
  __syncthreads();

  // Cross-wave reduction via WMMA (executed by ALL waves so EXEC is all-1s,
  // as required by the WMMA restrictions; redundant work is trivial).
  // A-matrix 16x4 f32 layout (ISA 7.12.2): lanes 0-15: VGPR0=K0, VGPR1=K1;
  // lanes 16-31: VGPR0=K2, VGPR1=K3. Row M = lane&15 = quantity index.
  const int q = lane & 15;
  v2f Aa, Ab;
  if (lane < 16) { Aa.x = wsum[0][q]; Aa.y = wsum[1][q];
                   Ab.x = wsum[4][q]; Ab.y = wsum[5][q]; }
  else           { Aa.x = wsum[2][q]; Aa.y = wsum[3][q];
                   Ab.x = wsum[6][q]; Ab.y = wsum[7][q]; }
  v2f Bones; Bones.x = 1.0f; Bones.y = 1.0f;   // ones => layout-independent

  v8f d = {};
  d = __builtin_amdgcn_wmma_f32_16x16x4_f32(false, Aa, false, Bones,
                                            (short)0, d, false, false);
  d = __builtin_amdgcn_wmma_f32_16x16x4_f32(false, Ab, false, Bones,
                                            (short)0, d, false, false);
  // D layout: C/D VGPR r on lanes 0-15 holds (M=r, N=lane): lane 0's element
  // r is the block total of quantity r.
  if (tid == 0) {
    float* o = part + ((size_t)bj * NCHUNK + blockIdx.y) * 8;
    o[0] = d[0]; o[1] = d[1]; o[2] = d[2]; o[3] = d[3];
    o[4] = d[4]; o[5] = d[5]; o[6] = d[6]; o[7] = d[7];
  }
}

// ---------------------------------------------------------------------------
// Kernel 2: fixed-order final sum of the per-chunk partials (deterministic).
// ---------------------------------------------------------------------------
__global__ void reduce_parts(const float* __restrict__ part,
                             float* __restrict__ acc) {
  int t = blockIdx.x * blockDim.x + threadIdx.x;   // 0..1023
  if (t >= NB * NJ * 8) return;
  int bj = t >> 3, q = t & 7;
  float s = 0.0f;
  for (int c = 0; c < NCHUNK; ++c)
    s += part[((size_t)bj * NCHUNK + c) * 8 + q];
  acc[t] = s;
}

// ---------------------------------------------------------------------------
// Kernel 3: per-(b,joint) Rodrigues + repose (stores 3x4 [R|t], row-major).
// Faithful to reference: angle = ||v + 1e-8|| (elementwise offset), axis = v/angle,
// padded joint (jj==NJ) uses locs=poses=0 through the same formula.
// ---------------------------------------------------------------------------
__global__ void repose_kernel(const float* __restrict__ acc,
                              float* __restrict__ rep) {
  int t = threadIdx.x;
  if (t >= NB * (NJ + 1)) return;
  int b  = t / (NJ + 1);
  int jj = t % (NJ + 1);

  float loc[3]  = {0.f, 0.f, 0.f};
  float pose[3] = {0.f, 0.f, 0.f};
  if (jj < NJ) {
    const float* a = acc + ((size_t)b * NJ + jj) * 8;
    float denom = a[0] + 1e-5f;
    float inv = 1.0f / denom;
    loc[0]  = a[1] * inv; loc[1]  = a[2] * inv; loc[2]  = a[3] * inv;
    pose[0] = a[4] * inv; pose[1] = a[5] * inv; pose[2] = a[6] * inv;
  }
  float vx = -pose[0], vy = -pose[1], vz = -pose[2];
  float ex = vx + 1e-8f, ey = vy + 1e-8f, ez = vz + 1e-8f;
  float angle = sqrtf(ex * ex + ey * ey + ez * ez);
  float inva = 1.0f / angle;
  float ax = vx * inva, ay = vy * inva, az = vz * inva;
  float c = cosf(angle), s = sinf(angle), omc = 1.0f - c;

  float R[3][3];
  R[0][0] = c + omc * ax * ax;      R[0][1] = -s * az + omc * ax * ay; R[0][2] =  s * ay + omc * ax * az;
  R[1][0] =  s * az + omc * ay * ax; R[1][1] = c + omc * ay * ay;      R[1][2] = -s * ax + omc * ay * az;
  R[2][0] = -s * ay + omc * az * ax; R[2][1] =  s * ax + omc * az * ay; R[2][2] = c + omc * az * az;

  float* o = rep + ((size_t)b * (NJ + 1) + jj) * 12;
  #pragma unroll
  for (int r = 0; r < 3; ++r) {
    float tr = loc[r] - (R[r][0] * loc[0] + R[r][1] * loc[1] + R[r][2] * loc[2]);
    o[r * 4 + 0] = R[r][0]; o[r * 4 + 1] = R[r][1];
    o[r * 4 + 2] = R[r][2]; o[r * 4 + 3] = tr;
  }
}

// ---------------------------------------------------------------------------
// Kernel 4: per-pixel apply: mask gate, skin argmax (first-max), matrix
// gather, matvec + translation, valid3 gate. Output layout (B,3,H,W).
// ---------------------------------------------------------------------------
__global__ void apply_kernel(const float* __restrict__ seg,
                             const float* __restrict__ rep,
                             float* __restrict__ out) {
  size_t idx = (size_t)blockIdx.x * blockDim.x + threadIdx.x;
  if (idx >= (size_t)NB * NP) return;
  int b = (int)(idx >> 16);       // NP = 65536
  int p = (int)(idx & (NP - 1));
  const float* sb = seg + (size_t)b * NCH * NP;

  float n0 = sb[p];
  float n1 = sb[NP + p];
  float n2 = sb[2 * NP + p];
  float m  = sigm(sb[(size_t)CH_MASK * NP + p]);
  bool valid = m > 0.75f;
  n0 = valid ? n0 : 0.0f;
  n1 = valid ? n1 : 0.0f;
  n2 = valid ? n2 : 0.0f;

  // argmax over 18 skin channels; strict '>' keeps first max like jnp.argmax
  const float* sk = sb + (size_t)CH_SKIN * NP;
  float best = sk[p];
  int bi = 0;
  #pragma unroll
  for (int q = 1; q < NJ + 2; ++q) {
    float v = sk[(size_t)q * NP + p];
    if (v > best) { best = v; bi = q; }
  }

  float o0 = 0.f, o1 = 0.f, o2 = 0.f;
  if (bi >= 1) {    // onehot matches arange(1, J+2) -> joint = bi-1
    const float* T = rep + ((size_t)b * (NJ + 1) + (bi - 1)) * 12;
    o0 = T[0] * n0 + T[1]  * n1 + T[2]  * n2 + T[3];
    o1 = T[4] * n0 + T[5]  * n1 + T[6]  * n2 + T[7];
    o2 = T[8] * n0 + T[9]  * n1 + T[10] * n2 + T[11];
  }
  bool v3 = (n0 > 0.f) && (n1 > 0.f) && (n2 > 0.f);
  if (!v3) { o0 = 0.f; o1 = 0.f; o2 = 0.f; }

  out[((size_t)b * 3 + 0) * NP + p] = o0;
  out[((size_t)b * 3 + 1) * NP + p] = o1;
  out[((size_t)b * 3 + 2) * NP + p] = o2;
}

// ---------------------------------------------------------------------------
extern "C" void kernel_launch(void* const* d_in, const int* in_sizes, int n_in,
                              void* d_out, int out_size, void* d_ws, size_t ws_size,
                              hipStream_t stream) {
  const float* seg = (const float*)d_in[0];
  float* out = (float*)d_out;

  // Workspace layout (floats):
  //   [0 .. 65536)            per-chunk partials: [bj][chunk][8]
  //   [65536 .. 66560)        reduced accumulators: [bj][8]
  //   [66560 .. 68192)        repose matrices: [b][17][12]
  float* ws_part = (float*)d_ws;
  float* ws_acc  = ws_part + (size_t)NB * NJ * NCHUNK * 8;
  float* ws_rep  = ws_acc + (size_t)NB * NJ * 8;

  vote_kernel<<<dim3(NB * NJ, NCHUNK), 256, 0, stream>>>(seg, ws_part);
  reduce_parts<<<4, 256, 0, stream>>>(ws_part, ws_acc);
  repose_kernel<<<1, 256, 0, stream>>>(ws_acc, ws_rep);
  apply_kernel<<<(NB * NP + 255) / 256, 256, 0, stream>>>(seg, ws_rep, out);
}